// ResidualVectorQuantize_33328946217032
// MI455X (gfx1250) — compile-verified
//
#include <hip/hip_runtime.h>
#include <hip/hip_bf16.h>

// CDNA5 (gfx1250) fused Residual Vector Quantize forward.
// One wave32 per block owns 16 timesteps; residual tile lives in LDS for all
// 9 codebook stages. GEMMs run on v_wmma_f32_16x16x4_f32 with A-operands
// pre-packed in WMMA lane order (coalesced b64 loads), B-operands from LDS.
// z tile in / z_q tile out move through the Tensor Data Mover (async DMA).

typedef float v2f  __attribute__((ext_vector_type(2)));
typedef float v8f  __attribute__((ext_vector_type(8)));
typedef unsigned int u32x4 __attribute__((ext_vector_type(4)));
typedef int   i32x4 __attribute__((ext_vector_type(4)));
typedef int   i32x8 __attribute__((ext_vector_type(8)));

#define WMMA4(A, Bm, Cm) \
  __builtin_amdgcn_wmma_f32_16x16x4_f32(false, (A), false, (Bm), (short)0, (Cm), false, false)

#define HAVE_TDM (__has_builtin(__builtin_amdgcn_tensor_load_to_lds) && \
                  __has_builtin(__builtin_amdgcn_tensor_store_from_lds))

namespace {
constexpr int kB = 16, kD = 512, kT = 4096;
constexpr int kN = 9, kK = 1024, kC = 8;
constexpr size_t OFF_CODES = (size_t)kB * kD * kT;                 // 33,554,432
constexpr size_t OFF_LAT   = OFF_CODES + (size_t)kB * kN * kT;     // 34,144,256
constexpr size_t OFF_LOSS  = OFF_LAT + (size_t)kB * kN * kC * kT;  // 38,862,848
constexpr float  kInvMean  = 1.0f / ((float)kB * kC * kT);
// d_ws layout (in float2 units): packed WMMA A-operands
constexpr size_t P_CBN = 0;                       // kN*4096 float2 (normalized codebooks)
constexpr size_t P_IW  = P_CBN + (size_t)kN*4096; // kN*4096 float2 (in_w, rows 8..15 zero)
constexpr size_t P_OW  = P_IW  + (size_t)kN*4096; // kN*2048 float2 (out_w)
}

// ---------------------------------------------------------------------------
// Packing kernels: emit A-operands in WMMA lane order.
// float2 index layout: ((tile*nch + ch)*2 + half)*16 + row  ->  pair of
// consecutive K elements (k = ch*4 + 2*half) for matrix row `tile*16+row`.
// ---------------------------------------------------------------------------
__global__ void rvq_pack_iw(const float* __restrict__ iw, float2* __restrict__ dst) {
  int o = blockIdx.x * 256 + threadIdx.x;
  if (o >= kN * 4096) return;
  int n = o >> 12, r = o & 4095;
  int m = r & 15, q = r >> 4, h = q & 1, kc = q >> 1;   // kc: 0..127 (K chunk of 4)
  float2 v; v.x = 0.f; v.y = 0.f;
  if (m < 8) {
    const float* src = iw + ((size_t)n * kC + m) * kD + kc * 4 + 2 * h;
    v.x = src[0]; v.y = src[1];
  }
  dst[o] = v;
}

__global__ void rvq_pack_cbn(const float* __restrict__ cb, float2* __restrict__ dst) {
  int o = blockIdx.x * 256 + threadIdx.x;
  if (o >= kN * 4096) return;
  int n = o >> 12, r = o & 4095;
  int m = r & 15, q = r >> 4, h = q & 1, p = q >> 1, ch = p & 1, rt = p >> 1;
  int row = rt * 16 + m;                                 // codebook entry 0..1023
  const float* src = cb + ((size_t)n * kK + row) * kC;
  float v[8], s = 0.f;
#pragma unroll
  for (int c = 0; c < 8; ++c) { v[c] = src[c]; s += v[c] * v[c]; }
  float inv = 1.0f / fmaxf(sqrtf(s), 1e-12f);
  int c0 = ch * 4 + 2 * h;
  float2 outv; outv.x = v[c0] * inv; outv.y = v[c0 + 1] * inv;
  dst[o] = outv;
}

__global__ void rvq_pack_ow(const float* __restrict__ ow, float2* __restrict__ dst) {
  int o = blockIdx.x * 256 + threadIdx.x;
  if (o >= kN * 2048) return;
  int n = o / 2048, r = o % 2048;
  int m = r & 15, q = r >> 4, h = q & 1, p = q >> 1, ch = p & 1, dt = p >> 1;
  int row = dt * 16 + m;                                 // output channel d
  const float* src = ow + ((size_t)n * kD + row) * kC;
  int c0 = ch * 4 + 2 * h;
  float2 outv; outv.x = src[c0]; outv.y = src[c0 + 1];
  dst[o] = outv;
}

// ---------------------------------------------------------------------------
// 2D TDM descriptor for a 512(d) x 16(t) f32 tile with row stride T, dense in
// LDS. `ga` = global byte address of tile origin, `lds` = LDS byte address.
// ---------------------------------------------------------------------------
#if HAVE_TDM
__device__ __forceinline__ void tdm_fill(unsigned long long ga, unsigned lds,
                                         u32x4& g0, i32x8& g1) {
  g0[0] = 1u;                                                 // count=1 valid
  g0[1] = lds;                                                // lds_addr
  g0[2] = (unsigned)(ga & 0xFFFFFFFFu);                       // global_addr[31:0]
  g0[3] = (unsigned)((ga >> 32) & 0x01FFFFFFu) | (2u << 30);  // addr[56:32]|type=2
  g1[0] = 0x00020000;                 // data_size = 4B
  g1[1] = (int)(4096u << 16);         // tensor_dim0 = T (low16)
  g1[2] = (int)(512u << 16);          // tensor_dim0 hi=0 | tensor_dim1 = 512
  g1[3] = (int)(16u << 16);           // tensor_dim1 hi=0 | tile_dim0 = 16
  g1[4] = 512;                        // tile_dim1 = 512, tile_dim2 = 0
  g1[5] = 4096;                       // tensor_dim0_stride = T (low32)
  g1[6] = 0;
  g1[7] = 0;
}
#endif

// ---------------------------------------------------------------------------
// Fused RVQ: grid = B * (T/16) blocks of 32 threads (one wave).
// ---------------------------------------------------------------------------
__launch_bounds__(32)
__global__ void rvq_fused(const float* __restrict__ z,
                          const float* __restrict__ in_b,
                          const float* __restrict__ out_b,
                          const float* __restrict__ codebooks,
                          const float2* __restrict__ ws,
                          float* __restrict__ out) {
  __shared__ float res_s[kD][16];   // residual tile, resident across all stages
  __shared__ float ze_s[kC][16];    // z_e staging (B operand of code search)
  __shared__ float zq_s[kC][16];    // gathered codes (B operand of out_proj)
  __shared__ float ob_s[kD];
  __shared__ float ib_s[kC];

  const int l    = threadIdx.x;     // 0..31
  const int half = l >> 4;
  const int lo   = l & 15;
  const int b    = blockIdx.x >> 8;
  const int t0   = (blockIdx.x & 255) << 4;
  const size_t tile_elem = ((size_t)b * kD) * kT + t0;   // tile origin in z / out

  // ---- bring z tile into res_s via the Tensor Data Mover ------------------
#if HAVE_TDM
  {
    u32x4 g0; i32x8 g1;
    tdm_fill((unsigned long long)(uintptr_t)(z + tile_elem),
             (unsigned)(uintptr_t)&res_s[0][0], g0, g1);
    i32x4 gz4 = {0, 0, 0, 0};
    i32x8 gz8 = {0, 0, 0, 0, 0, 0, 0, 0};
    __builtin_amdgcn_tensor_load_to_lds(g0, g1, gz4, gz4, gz8, 0);
    __builtin_amdgcn_s_wait_tensorcnt(0);
  }
#else
  {
    const int tt = (l & 3) << 2;
    const int dr = l >> 2;
    for (int j = 0; j < 64; ++j) {
      int d = j * 8 + dr;
      float4 v = *(const float4*)(z + tile_elem + (size_t)d * kT + tt);
      *(float4*)&res_s[d][tt] = v;
    }
  }
#endif
  __syncthreads();

  float loss_acc = 0.f;

  for (int n = 0; n < kN; ++n) {
    const float2* awp = ws + P_IW  + (size_t)n * 4096;  // in_proj A pack
    const float2* cwp = ws + P_CBN + (size_t)n * 4096;  // codebook A pack (normalized)
    const float2* owp = ws + P_OW  + (size_t)n * 2048;  // out_proj A pack
    if (n + 1 < kN) {
      __builtin_prefetch(awp + 4096, 0, 1);
      __builtin_prefetch(cwp + 4096, 0, 1);
      __builtin_prefetch(owp + 2048, 0, 1);
    }
    if (l < 8) ib_s[l] = in_b[n * kC + l];
    for (int j = l; j < kD; j += 32) ob_s[j] = out_b[n * kD + j];
    __syncthreads();

    // ---- in_proj: z_e[c][t] = sum_d iw[c][d]*res[d][t]  (M=16 pad, K=512) --
    v8f a0 = {0,0,0,0,0,0,0,0}, a1 = {0,0,0,0,0,0,0,0};
    v8f a2 = {0,0,0,0,0,0,0,0}, a3 = {0,0,0,0,0,0,0,0};
    auto mm_in = [&](int kc, v8f cacc) -> v8f {
      float2 av = awp[(kc * 2 + half) * 16 + lo];        // coalesced 256B/wave
      v2f a; a.x = av.x; a.y = av.y;
      int kb = kc * 4 + (half << 1);
      v2f bb; bb.x = res_s[kb][lo]; bb.y = res_s[kb + 1][lo];
      return WMMA4(a, bb, cacc);
    };
    for (int kc = 0; kc < 128; kc += 4) {                // 4 chains for WMMA ILP
      a0 = mm_in(kc,     a0);
      a1 = mm_in(kc + 1, a1);
      a2 = mm_in(kc + 2, a2);
      a3 = mm_in(kc + 3, a3);
    }
    v8f acc = (a0 + a1) + (a2 + a3);

    // lanes 0..15 hold z_e rows c=0..7 for t=lane: bias, stage, emit latents.
    __syncthreads();
    if (l < 16) {
      float* latp = out + OFF_LAT + ((size_t)(b * (kN * kC) + n * kC)) * kT + t0 + l;
#pragma unroll
      for (int c = 0; c < 8; ++c) {
        float v = acc[c] + ib_s[c];
        ze_s[c][l] = v;
        latp[(size_t)c * kT] = v;
      }
    }
    __syncthreads();

    // ---- code search: argmax_k <z_e, cbn_k> (all norms constant) -----------
    float best = -3.4e38f;
    int   bidx = 0;
    for (int rt = 0; rt < kK / 16; ++rt) {
      v8f dacc = {0,0,0,0,0,0,0,0};
#pragma unroll
      for (int ch = 0; ch < 2; ++ch) {
        float2 av = cwp[((rt * 2 + ch) * 2 + half) * 16 + lo];
        v2f a; a.x = av.x; a.y = av.y;
        int kb = ch * 4 + (half << 1);
        v2f bb; bb.x = ze_s[kb][lo]; bb.y = ze_s[kb + 1][lo];
        dacc = WMMA4(a, bb, dacc);
      }
#pragma unroll
      for (int i = 0; i < 8; ++i) {
        int kv = rt * 16 + i + half * 8;
        float v = dacc[i];
        if (v > best) { best = v; bidx = kv; }
      }
    }
    {  // merge lane halves (same t, disjoint code rows)
      float v2 = __shfl(best, lo + 16);
      int   i2 = __shfl(bidx, lo + 16);
      if (v2 > best || (v2 == best && i2 < bidx)) { best = v2; bidx = i2; }
    }

    // ---- gather unnormalized code row, losses, codes output ----------------
    if (l < 16) {
      const float* cbrow = codebooks + ((size_t)n * kK + bidx) * kC;
      float4 q0 = *(const float4*)cbrow;
      float4 q1 = *(const float4*)(cbrow + 4);
      float qs[8] = {q0.x, q0.y, q0.z, q0.w, q1.x, q1.y, q1.z, q1.w};
      float ls = 0.f;
#pragma unroll
      for (int c = 0; c < 8; ++c) {
        zq_s[c][l] = qs[c];
        float d = ze_s[c][l] - qs[c];
        ls += d * d;
      }
      loss_acc += ls;
      out[OFF_CODES + ((size_t)(b * kN + n)) * kT + t0 + l] = (float)bidx;
    }
    __syncthreads();

    // ---- out_proj + residual update ---------------------------------------
    for (int dt = 0; dt < kD / 16; ++dt) {
      v8f oacc = {0,0,0,0,0,0,0,0};
#pragma unroll
      for (int ch = 0; ch < 2; ++ch) {
        float2 av = owp[((dt * 2 + ch) * 2 + half) * 16 + lo];
        v2f a; a.x = av.x; a.y = av.y;
        int kb = ch * 4 + (half << 1);
        v2f bb; bb.x = zq_s[kb][lo]; bb.y = zq_s[kb + 1][lo];
        oacc = WMMA4(a, bb, oacc);
      }
#pragma unroll
      for (int i = 0; i < 8; ++i) {
        int d = dt * 16 + i + half * 8;
        float v = oacc[i] + ob_s[d];
        res_s[d][lo] -= v;
      }
    }
    __syncthreads();
  }

  // ---- z_q output: zq_acc = z - final residual ------------------------------
  // Compute in place in LDS, then DMA the tile out with the TDM.
  {
    const int tt = (l & 3) << 2;
    const int dr = l >> 2;
    for (int j = 0; j < 64; ++j) {
      int d = j * 8 + dr;
      float4 zv = *(const float4*)(z + tile_elem + (size_t)d * kT + tt);
      float4 rv = *(const float4*)&res_s[d][tt];
      float4 o;
      o.x = zv.x - rv.x; o.y = zv.y - rv.y;
      o.z = zv.z - rv.z; o.w = zv.w - rv.w;
#if HAVE_TDM
      *(float4*)&res_s[d][tt] = o;                   // stage z_q tile in LDS
#else
      *(float4*)(out + tile_elem + (size_t)d * kT + tt) = o;
#endif
    }
  }
#if HAVE_TDM
  __syncthreads();
  asm volatile("s_wait_dscnt 0x0" ::: "memory");     // TDM is unordered vs LDS ops
  {
    u32x4 g0; i32x8 g1;
    tdm_fill((unsigned long long)(uintptr_t)(out + tile_elem),
             (unsigned)(uintptr_t)&res_s[0][0], g0, g1);
    i32x4 gz4 = {0, 0, 0, 0};
    i32x8 gz8 = {0, 0, 0, 0, 0, 0, 0, 0};
    __builtin_amdgcn_tensor_store_from_lds(g0, g1, gz4, gz4, gz8, 0);
  }
#endif

  // ---- loss scalars (commit == cb_loss in eval mode) ------------------------
#pragma unroll
  for (int off = 16; off >= 1; off >>= 1) loss_acc += __shfl_xor(loss_acc, off);
  if (l == 0) {
    float v = loss_acc * kInvMean;
    atomicAdd(out + OFF_LOSS, v);
    atomicAdd(out + OFF_LOSS + 1, v);
  }
#if HAVE_TDM
  __builtin_amdgcn_s_wait_tensorcnt(0);              // tile flushed before wave end
#endif
}

// ---------------------------------------------------------------------------
extern "C" void kernel_launch(void* const* d_in, const int* in_sizes, int n_in,
                              void* d_out, int out_size, void* d_ws, size_t ws_size,
                              hipStream_t stream) {
  (void)in_sizes; (void)n_in; (void)out_size; (void)ws_size;
  const float* z         = (const float*)d_in[0];
  const float* in_w      = (const float*)d_in[1];
  const float* in_b      = (const float*)d_in[2];
  const float* out_w     = (const float*)d_in[3];
  const float* out_b     = (const float*)d_in[4];
  const float* codebooks = (const float*)d_in[5];
  float*  out = (float*)d_out;
  float2* ws  = (float2*)d_ws;   // 92,160 float2 = 737,280 B packed A-operands

  (void)hipMemsetAsync((void*)(out + OFF_LOSS), 0, 2 * sizeof(float), stream);

  rvq_pack_cbn<<<(kN * 4096 + 255) / 256, 256, 0, stream>>>(codebooks, ws + P_CBN);
  rvq_pack_iw <<<(kN * 4096 + 255) / 256, 256, 0, stream>>>(in_w,      ws + P_IW);
  rvq_pack_ow <<<(kN * 2048 + 255) / 256, 256, 0, stream>>>(out_w,     ws + P_OW);

  dim3 grid(kB * (kT / 16));  // 4096 blocks, 1 wave each
  rvq_fused<<<grid, 32, 0, stream>>>(z, in_b, out_b, codebooks, ws, out);
}